// CSOCSSC_V124_72739566125272
// MI455X (gfx1250) — compile-verified
//
#include <hip/hip_runtime.h>

typedef __attribute__((ext_vector_type(2))) float v2f;
typedef __attribute__((ext_vector_type(8))) float v8f;

#define NPTS 8192
#define DIMC 128
#define NSEG 4
#define SEGLEN (NPTS / NSEG)
// pair-interleaved LDS pitch in float2 units; 2*144 mod 64 == 32 -> lanes 0-15
// and 16-31 (adjacent kp rows) hit disjoint bank halves on b64 loads
#define PITCH2 144

// -(log2 e)/12
#define NEG_LOG2E_OVER_12 (-0.12022458674074695f)

// ---------------- Pass 1: unnormalized K + row sums ----------------
__global__ __launch_bounds__(256) void pair_kernel(const float* __restrict__ coords,
                                                   const float* __restrict__ alpha,
                                                   float* __restrict__ Kout,
                                                   float* __restrict__ rowsum) {
  const int i   = blockIdx.x;
  const int tid = threadIdx.x;

  const float cx = coords[3 * i + 0];
  const float cy = coords[3 * i + 1];
  const float cz = coords[3 * i + 2];
  const float sqi = cx * cx + cy * cy + cz * cz;
  const float ai  = alpha[i];

  float* __restrict__ krow = Kout + (size_t)i * NPTS;

  float local = 0.0f;
  for (int j = tid; j < NPTS; j += 256) {
    const float bx = coords[3 * j + 0];
    const float by = coords[3 * j + 1];
    const float bz = coords[3 * j + 2];
    float d2 = sqi + (bx * bx + by * by + bz * bz)
             - 2.0f * (cx * bx + cy * by + cz * bz);
    float D = __builtin_amdgcn_sqrtf(fmaxf(d2, 1e-12f)) + 1e-6f;
    float a = 0.5f * (ai + alpha[j]);
    // D^(-a) = exp2(-a * log2(D)); D >= 1e-6 so the ref's clip is a no-op
    float p = __builtin_amdgcn_exp2f(-a * __builtin_amdgcn_logf(D));
    p = fminf(fmaxf(p, 1e-8f), 1000.0f);
    float K = p * __builtin_amdgcn_exp2f(D * NEG_LOG2E_OVER_12);
    K = fminf(fmaxf(K, 0.0f), 1000.0f);
    if (j == i) K = 0.0f;
    krow[j] = K;
    local += K;
  }

  __shared__ float red[256];
  red[tid] = local;
  __syncthreads();
  #pragma unroll
  for (int s = 128; s > 0; s >>= 1) {
    if (tid < s) red[tid] += red[tid + s];
    __syncthreads();
  }
  if (tid == 0) rowsum[i] = red[0];
}

// ---- Pass 2: normalize K in place + partial out = K_seg @ latent_seg (WMMA fp32) ----
// grid = (128, S). Block (rb, s): rows rb*64..+63, j in [s*seglen, (s+1)*seglen).
// Partial result stored to outbuf + s * NPTS*DIMC (plain stores, deterministic).
__global__ __launch_bounds__(128) void agg_kernel(const float* __restrict__ latent,
                                                  const float* __restrict__ rowsum,
                                                  float* __restrict__ Kmat,
                                                  float* __restrict__ outbuf,
                                                  int seglen) {
  // pair-interleaved latent chunk: lt[kp*2*PITCH2 + 2*col] = latent[2kp][col],
  //                               lt[kp*2*PITCH2 + 2*col + 1] = latent[2kp+1][col]
  __shared__ __align__(16) float lt[64 * 2 * PITCH2];  // 72 KB

  const int tid  = threadIdx.x;
  const int wave = tid >> 5;
  const int lane = tid & 31;
  const int l16  = lane & 15;
  const int hi   = lane >> 4;     // 0 for lanes 0-15, 1 for lanes 16-31
  const int ksel = hi << 1;       // A operand k-offset: 0 or 2

  const int rowbase = blockIdx.x * 64 + wave * 16;
  const int jbase   = blockIdx.y * seglen;

  const float s    = rowsum[rowbase + l16];
  const float rinv = 1.0f / (s + 1e-8f);

  // this lane's K row, pre-offset by its A-layout k-select
  float* __restrict__ krow = Kmat + (size_t)(rowbase + l16) * NPTS + ksel;

  const v8f zero = {0.f, 0.f, 0.f, 0.f, 0.f, 0.f, 0.f, 0.f};
  v8f acc[8];
  #pragma unroll
  for (int t = 0; t < 8; ++t) acc[t] = zero;

  const int bcol = l16 << 1;  // float2-interleaved column offset for this lane

  for (int j0 = jbase; j0 < jbase + seglen; j0 += 128) {
    __syncthreads();
    // stage latent[j0..j0+127][:] pair-interleaved: 4096 float4 tasks / 128 thr
    const float* __restrict__ srcbase = latent + (size_t)j0 * DIMC;
    #pragma unroll
    for (int k = 0; k < 32; ++k) {
      const int q  = k * 128 + tid;     // 0..4095
      const int kp = q >> 6;            // row pair 0..63
      const int c0 = (q & 63) << 1;     // even column 0..126
      const float* r0 = srcbase + (size_t)(kp << 1) * DIMC + c0;
      float2 a = *reinterpret_cast<const float2*>(r0);
      float2 b = *reinterpret_cast<const float2*>(r0 + DIMC);
      float4 w;
      w.x = a.x; w.y = b.x; w.z = a.y; w.w = b.y;
      *reinterpret_cast<float4*>(lt + 2 * (kp * PITCH2 + c0)) = w;
    }
    __syncthreads();

    for (int jj = 0; jj < 128; jj += 4) {
      // A (16x4 f32): lanes 0-15 carry k = jj,jj+1 ; lanes 16-31 carry jj+2,jj+3
      float2 ka = *reinterpret_cast<const float2*>(krow + j0 + jj);
      float2 kn;
      kn.x = ka.x * rinv;
      kn.y = ka.y * rinv;
      // write normalized K back in place (second reference output)
      *reinterpret_cast<float2*>(krow + j0 + jj) = kn;
      v2f A;
      A.x = kn.x;
      A.y = kn.y;

      // base of this lane's k-pair row in interleaved LDS
      const int kp   = (jj + ksel) >> 1;           // even jj+ksel -> exact pair
      const int base = kp * (2 * PITCH2) + bcol;
      #pragma unroll
      for (int t = 0; t < 8; ++t) {
        // one aligned b64 LDS load yields {k even, k odd} for this column
        v2f B = *reinterpret_cast<const v2f*>(lt + base + (t << 5));
        acc[t] = __builtin_amdgcn_wmma_f32_16x16x4_f32(
            /*neg_a=*/false, A, /*neg_b=*/false, B,
            /*c_mod=*/(short)0, acc[t], /*reuse_a=*/false, /*reuse_b=*/false);
      }
    }
  }

  // D layout (16x16 f32): VGPR r -> (M=r, N=lane) lanes 0-15, (M=r+8, N=lane-16) lanes 16-31
  float* __restrict__ od = outbuf + (size_t)blockIdx.y * ((size_t)NPTS * DIMC);
  #pragma unroll
  for (int t = 0; t < 8; ++t) {
    #pragma unroll
    for (int r = 0; r < 8; ++r) {
      const int row = rowbase + r + (hi << 3);
      const int col = t * 16 + l16;
      od[(size_t)row * DIMC + col] = acc[t][r];
    }
  }
}

// ---------------- Pass 3: sum NSEG partials into out ----------------
__global__ __launch_bounds__(256) void reduce_kernel(const float* __restrict__ part,
                                                     float* __restrict__ outp) {
  const size_t n   = (size_t)NPTS * DIMC;
  const size_t idx = (size_t)blockIdx.x * 256 + threadIdx.x;
  float v = part[idx] + part[idx + n] + part[idx + 2 * n] + part[idx + 3 * n];
  outp[idx] = v;
}

extern "C" void kernel_launch(void* const* d_in, const int* in_sizes, int n_in,
                              void* d_out, int out_size, void* d_ws, size_t ws_size,
                              hipStream_t stream) {
  const float* latent = (const float*)d_in[0];  // [8192,128]
  const float* coords = (const float*)d_in[1];  // [8192,3]
  const float* alpha  = (const float*)d_in[2];  // [8192]

  float* outp = (float*)d_out;                  // [8192,128]
  float* Kmat = outp + (size_t)NPTS * DIMC;     // [8192,8192]
  float* rowsum = (float*)d_ws;                 // 8192 floats of scratch

  pair_kernel<<<NPTS, 256, 0, stream>>>(coords, alpha, Kmat, rowsum);

  const size_t need = (size_t)NPTS * sizeof(float)
                    + (size_t)NSEG * NPTS * DIMC * sizeof(float);
  if (ws_size >= need) {
    // segmented path: 512 blocks / 2048 waves for latency hiding
    float* part = rowsum + NPTS;
    dim3 grid(NPTS / 64, NSEG);
    agg_kernel<<<grid, 128, 0, stream>>>(latent, rowsum, Kmat, part, SEGLEN);
    reduce_kernel<<<(NPTS * DIMC) / 256, 256, 0, stream>>>(part, outp);
  } else {
    // fallback: single segment writes out directly
    dim3 grid(NPTS / 64, 1);
    agg_kernel<<<grid, 128, 0, stream>>>(latent, rowsum, Kmat, outp, NPTS);
  }
}